// PrefetchDenseInstanceNorm_64158221467882
// MI455X (gfx1250) — compile-verified
//
#include <hip/hip_runtime.h>
#include <hip/hip_bf16.h>

// Fixed problem shapes from the reference
#define C_CH 128
#define H_DIM 256
#define W_DIM 256
#define CHW (C_CH * H_DIM * W_DIM)   // 8,388,608 elements per half
#define TBL 16                        // anchor table is 16x16

typedef __attribute__((ext_vector_type(2))) float v2f;
typedef __attribute__((ext_vector_type(8))) float v8f;

// ---------------------------------------------------------------------------
// Kernel 1: per-channel mean / unbiased std of the "pre" half.
// One block per channel, fixed-order tree reduction => bitwise deterministic.
// ---------------------------------------------------------------------------
__global__ void pdin_stats_kernel(const float* __restrict__ x,
                                  float* __restrict__ pre_mean,
                                  float* __restrict__ pre_std,
                                  float* __restrict__ pre_stdinv) {
    const int c = blockIdx.x;
    const float* __restrict__ p = x + CHW + (size_t)c * (H_DIM * W_DIM);
    const int N  = H_DIM * W_DIM;        // 65536
    const int N4 = N >> 2;               // float4 count

    float s = 0.0f, sq = 0.0f;
    const float4* p4 = reinterpret_cast<const float4*>(p);
    for (int i = threadIdx.x; i < N4; i += blockDim.x) {
        float4 v = p4[i];
        s  += v.x + v.y + v.z + v.w;
        sq += v.x * v.x + v.y * v.y + v.z * v.z + v.w * v.w;
    }

    __shared__ float ssum[256];
    __shared__ float ssq[256];
    ssum[threadIdx.x] = s;
    ssq[threadIdx.x]  = sq;
    __syncthreads();
    for (int off = 128; off > 0; off >>= 1) {
        if (threadIdx.x < off) {
            ssum[threadIdx.x] += ssum[threadIdx.x + off];
            ssq[threadIdx.x]  += ssq[threadIdx.x + off];
        }
        __syncthreads();
    }
    if (threadIdx.x == 0) {
        float mean = ssum[0] / (float)N;
        float var  = (ssq[0] - (float)N * mean * mean) / (float)(N - 1);
        float sd   = sqrtf(var);
        pre_mean[c]   = mean;
        pre_std[c]    = sd;
        pre_stdinv[c] = 1.0f / sd;
    }
}

// ---------------------------------------------------------------------------
// Kernel 2 (tiny, 1 block): bicubic weight tables + 4x4 anchor patches.
// ---------------------------------------------------------------------------
__device__ __forceinline__ float pdin_cubic(float t) {
    const float a = -0.75f;
    float at = fabsf(t);
    float near = ((a + 2.0f) * at - (a + 3.0f)) * at * at + 1.0f;
    float far  = (((at - 5.0f) * at + 8.0f) * at - 4.0f) * a;
    return (at <= 1.0f) ? near : ((at < 2.0f) ? far : 0.0f);
}

// Effective 4-tap weights for output row `o` of the h/2..h/2+h crop of the
// (3h x 4) row-stochastic bicubic matrix, with index clamping folded in.
__device__ __forceinline__ void pdin_wrow(int o, int dim, float* eff) {
    float scale = 4.0f / (3.0f * (float)dim);
    float src = ((float)(dim / 2 + o) + 0.5f) * scale - 0.5f;
    float i0f = floorf(src);
    float t = src - i0f;
    int i0 = (int)i0f;
    eff[0] = eff[1] = eff[2] = eff[3] = 0.0f;
    #pragma unroll
    for (int k = 0; k < 4; ++k) {
        int off = k - 1;
        float wk = pdin_cubic(t - (float)off);
        int idx = i0 + off;
        idx = idx < 0 ? 0 : (idx > 3 ? 3 : idx);
        eff[idx] += wk;
    }
}

__global__ void pdin_setup_kernel(const float* __restrict__ mean_table,
                                  const float* __restrict__ std_table,
                                  const int* __restrict__ ya_p, const int* __restrict__ xa_p,
                                  const int* __restrict__ pya_p, const int* __restrict__ pxa_p,
                                  const float* __restrict__ pre_mean,
                                  const float* __restrict__ pre_std,
                                  float* __restrict__ Wy, float* __restrict__ Wx,
                                  float* __restrict__ mpatch, float* __restrict__ spatch) {
    const int tid = threadIdx.x;

    // Weight tables (H_DIM rows x 4 taps each)
    if (tid < H_DIM) {
        float e[4];
        pdin_wrow(tid, H_DIM, e);
        #pragma unroll
        for (int k = 0; k < 4; ++k) Wy[tid * 4 + k] = e[k];
        pdin_wrow(tid, W_DIM, e);
        #pragma unroll
        for (int k = 0; k < 4; ++k) Wx[tid * 4 + k] = e[k];
    }

    // 4x4xC patches with replication-pad clamp + table override
    int ya = *ya_p, xa = *xa_p, py = *pya_p, px = *pxa_p;
    ya = ya < 0 ? 0 : (ya > TBL - 1 ? TBL - 1 : ya);   // dynamic_slice start clamp
    xa = xa < 0 ? 0 : (xa > TBL - 1 ? TBL - 1 : xa);
    for (int i = tid; i < 4 * 4 * C_CH; i += blockDim.x) {
        int y = i / (4 * C_CH);
        int r = i % (4 * C_CH);
        int xx = r / C_CH;
        int c = r % C_CH;
        int iy = ya + y - 1; iy = iy < 0 ? 0 : (iy > TBL - 1 ? TBL - 1 : iy);
        int ix = xa + xx - 1; ix = ix < 0 ? 0 : (ix > TBL - 1 ? TBL - 1 : ix);
        float m, sd;
        if (iy == py && ix == px) { m = pre_mean[c]; sd = pre_std[c]; }
        else {
            int t = (iy * TBL + ix) * C_CH + c;
            m = mean_table[t]; sd = std_table[t];
        }
        mpatch[i] = m;
        spatch[i] = 1.0f / sd;
    }
}

// ---------------------------------------------------------------------------
// Kernel 3: fused bicubic-map (WMMA f32 16x16x4) + normalize of both halves.
// grid = (C, 32), block = 256 (8 waves); each wave owns one 16x16 tile.
// ---------------------------------------------------------------------------
__global__ void pdin_main_kernel(const float* __restrict__ x,
                                 const float* __restrict__ Wy,
                                 const float* __restrict__ Wx,
                                 const float* __restrict__ mpatch,
                                 const float* __restrict__ spatch,
                                 const float* __restrict__ pre_mean,
                                 const float* __restrict__ pre_stdinv,
                                 const float* __restrict__ weight,
                                 const float* __restrict__ bias,
                                 float* __restrict__ out) {
    const int c = blockIdx.x;

    // Stage this channel's 4x4 patches in LDS (32 floats total).
    __shared__ float lm[16];
    __shared__ float ls[16];
    if (threadIdx.x < 16)       lm[threadIdx.x]      = mpatch[threadIdx.x * C_CH + c];
    else if (threadIdx.x < 32)  ls[threadIdx.x - 16] = spatch[(threadIdx.x - 16) * C_CH + c];
    __syncthreads();

    const int wave = threadIdx.x >> 5;
    const int lane = threadIdx.x & 31;
    const int tile = blockIdx.y * 8 + wave;        // 0..255
    const int o0 = (tile >> 4) * 16;               // tile row origin
    const int p0 = (tile & 15) * 16;               // tile col origin

    const int half = lane >> 4;                    // 0: lanes 0-15, 1: lanes 16-31
    const int l15  = lane & 15;
    const int kb   = half * 2;                     // K pair held by this lane

    // A fragment: Wy tile rows (M = l15), taps K = kb, kb+1
    v2f a;
    a.x = Wy[(o0 + l15) * 4 + kb];
    a.y = Wy[(o0 + l15) * 4 + kb + 1];

    // B fragments: T[y][p] = sum_x patch[y][x] * Wx[p][x] for this lane's column
    const float wx0 = Wx[(p0 + l15) * 4 + 0];
    const float wx1 = Wx[(p0 + l15) * 4 + 1];
    const float wx2 = Wx[(p0 + l15) * 4 + 2];
    const float wx3 = Wx[(p0 + l15) * 4 + 3];
    v2f bm, bs;
    bm.x = lm[kb * 4 + 0] * wx0 + lm[kb * 4 + 1] * wx1 + lm[kb * 4 + 2] * wx2 + lm[kb * 4 + 3] * wx3;
    bm.y = lm[(kb + 1) * 4 + 0] * wx0 + lm[(kb + 1) * 4 + 1] * wx1 + lm[(kb + 1) * 4 + 2] * wx2 + lm[(kb + 1) * 4 + 3] * wx3;
    bs.x = ls[kb * 4 + 0] * wx0 + ls[kb * 4 + 1] * wx1 + ls[kb * 4 + 2] * wx2 + ls[kb * 4 + 3] * wx3;
    bs.y = ls[(kb + 1) * 4 + 0] * wx0 + ls[(kb + 1) * 4 + 1] * wx1 + ls[(kb + 1) * 4 + 2] * wx2 + ls[(kb + 1) * 4 + 3] * wx3;

    // D = A x B + 0 : mean_map tile and stdinv_map tile (A reused)
    v8f zero = {};
    v8f dm = __builtin_amdgcn_wmma_f32_16x16x4_f32(false, a, false, bm, (short)0, zero, false, false);
    v8f ds = __builtin_amdgcn_wmma_f32_16x16x4_f32(false, a, false, bs, (short)0, zero, false, false);

    const float wgt = weight[c];
    const float bi  = bias[c];
    const float pm  = pre_mean[c];
    const float psi = pre_stdinv[c];

    const size_t cbase = (size_t)c * (H_DIM * W_DIM);
    #pragma unroll
    for (int r = 0; r < 8; ++r) {
        const int row = o0 + half * 8 + r;         // C/D layout: VGPR r -> M = r (+8 hi lanes)
        const int col = p0 + l15;                  // lane -> N
        const size_t idx = cbase + (size_t)row * W_DIM + col;
        float xr = x[idx];
        out[idx] = (xr - dm[r]) * ds[r] * wgt + bi;
        float xp = x[CHW + idx];
        out[CHW + idx] = (xp - pm) * psi * wgt + bi;
    }
}

// ---------------------------------------------------------------------------
extern "C" void kernel_launch(void* const* d_in, const int* in_sizes, int n_in,
                              void* d_out, int out_size, void* d_ws, size_t ws_size,
                              hipStream_t stream) {
    const float* x          = (const float*)d_in[0];
    const float* mean_table = (const float*)d_in[1];
    const float* std_table  = (const float*)d_in[2];
    const float* weight     = (const float*)d_in[3];
    const float* bias       = (const float*)d_in[4];
    const int*   y_anchor   = (const int*)d_in[5];
    const int*   x_anchor   = (const int*)d_in[6];
    const int*   pre_y1     = (const int*)d_in[7];
    const int*   pre_x1     = (const int*)d_in[8];
    float* out = (float*)d_out;

    // Workspace layout (floats)
    float* w = (float*)d_ws;
    float* pre_mean   = w;            // 128
    float* pre_std    = w + 128;      // 128
    float* pre_stdinv = w + 256;      // 128
    float* Wy         = w + 384;      // 256*4
    float* Wx         = w + 1408;     // 256*4
    float* mpatch     = w + 2432;     // 4*4*128
    float* spatch     = w + 4480;     // 4*4*128   (total 6528 floats)

    pdin_stats_kernel<<<C_CH, 256, 0, stream>>>(x, pre_mean, pre_std, pre_stdinv);

    pdin_setup_kernel<<<1, 256, 0, stream>>>(mean_table, std_table,
                                             y_anchor, x_anchor, pre_y1, pre_x1,
                                             pre_mean, pre_std,
                                             Wy, Wx, mpatch, spatch);

    dim3 grid(C_CH, 32);
    pdin_main_kernel<<<grid, 256, 0, stream>>>(x, Wy, Wx, mpatch, spatch,
                                               pre_mean, pre_stdinv,
                                               weight, bias, out);
}